// KGAT_89258010346032
// MI455X (gfx1250) — compile-verified
//
#include <hip/hip_runtime.h>
#include <math.h>

typedef __attribute__((ext_vector_type(2))) float v2f;
typedef __attribute__((ext_vector_type(8))) float v8f;

#define EPS 1e-6f
#define MAX_NORM 0.999f

// ---------------- wave32 helpers ----------------
__device__ inline float wsum32(float v) {
#pragma unroll
  for (int o = 16; o > 0; o >>= 1) v += __shfl_xor(v, o, 32);
  return v;
}
// lane holds elements [2*lane, 2*lane+1] of a 64-vector
__device__ inline float dot64(float2 a, float2 b) {
  return wsum32(a.x * b.x + a.y * b.y);
}

// mobius_add with all reductions precomputed (x2=|x|^2, y2=|y|^2, xy=x.y)
__device__ inline float2 mobius_pre(float2 x, float2 y, float x2, float y2,
                                    float xy) {
  float ca = 1.f + 2.f * xy + y2;
  float cb = 1.f - x2;
  float inv = 1.f / fmaxf(1.f + 2.f * xy + x2 * y2, EPS);
  float2 r;
  r.x = (ca * x.x + cb * y.x) * inv;
  r.y = (ca * x.y + cb * y.y) * inv;
  return r;
}

// ---------------- kernels ----------------
__global__ void zero_kernel(float* p, size_t n) {
  size_t i = (size_t)blockIdx.x * blockDim.x + threadIdx.x;
  size_t st = (size_t)gridDim.x * blockDim.x;
  for (; i < n; i += st) p[i] = 0.f;
}

// One wave per edge (grid-stride). Hyperbolic tangent-space message with
// 11 wave-reductions per edge (redundant norms derived algebraically).
// Atomic segment-sum into d_out[0 : N*64], count into cnt1.
__global__ void kg_edge_kernel(const float* __restrict__ ent,
                               const float* __restrict__ rel_table,
                               const int* __restrict__ esrc,
                               const int* __restrict__ edst,
                               const int* __restrict__ etype,
                               float* __restrict__ out_sum,
                               float* __restrict__ cnt1, int E) {
  int lane = threadIdx.x & 31;
  int wave = ((int)(blockIdx.x * blockDim.x + threadIdx.x)) >> 5;
  int nwaves = (int)((gridDim.x * blockDim.x) >> 5);
  for (int e = wave; e < E; e += nwaves) {
    int s = esrc[e];
    int d = edst[e];
    int t = etype[e] + 2;
    float2 se = ((const float2*)(ent + (size_t)s * 64))[lane];
    float2 de = ((const float2*)(ent + (size_t)d * 64))[lane];
    float2 re = ((const float2*)(rel_table + (size_t)t * 64))[lane];

    // base = expmap0(se) = project(tanh(n)*se/n)      [reduction 1: s2]
    float s2 = dot64(se, se);
    float ns = sqrtf(fmaxf(s2, EPS));
    float sc0 = tanhf(ns) / ns;
    float t2 = sc0 * sc0 * s2;  // |tanh(n)*se/n|^2 (exact algebra)
    float tn = sqrtf(fmaxf(t2, EPS));
    float ps = (tn > MAX_NORM) ? (MAX_NORM / tn) : 1.f;
    float2 base;
    base.x = se.x * sc0 * ps;
    base.y = se.y * sc0 * ps;
    float b2 = t2 * ps * ps;  // |base|^2
    float lam = 2.f / fmaxf(1.f - b2, EPS);

    // a = expmap(de, base)                            [reductions 2,3]
    float d2 = dot64(de, de);
    float dn = sqrtf(fmaxf(d2, EPS));
    float sd = tanhf(0.5f * lam * dn) / dn;
    float2 secd;
    secd.x = de.x * sd;
    secd.y = de.y * sd;
    float secd2 = sd * sd * d2;
    float bd = sd * dot64(base, de);
    float2 a = mobius_pre(base, secd, b2, secd2, bd);

    // bvec = expmap(re, base)                         [reductions 4,5]
    float r2 = dot64(re, re);
    float rn = sqrtf(fmaxf(r2, EPS));
    float sr = tanhf(0.5f * lam * rn) / rn;
    float2 secr;
    secr.x = re.x * sr;
    secr.y = re.y * sr;
    float secr2 = sr * sr * r2;
    float br = sr * dot64(base, re);
    float2 bvec = mobius_pre(base, secr, b2, secr2, br);

    // msum = mobius_add(a, bvec)                      [reductions 6,7,8]
    float a2 = dot64(a, a);
    float bb2 = dot64(bvec, bvec);
    float ab = dot64(a, bvec);
    float2 msum = mobius_pre(a, bvec, a2, bb2, ab);

    // m = project(msum)                               [reduction 9]
    float m2 = dot64(msum, msum);
    float mn = sqrtf(fmaxf(m2, EPS));
    float mps = (mn > MAX_NORM) ? (MAX_NORM / mn) : 1.f;
    float2 m;
    m.x = msum.x * mps;
    m.y = msum.y * mps;
    float mm2 = m2 * mps * mps;

    // ts = logmap(m, base)                            [reductions 10,11]
    float2 nb;
    nb.x = -base.x;
    nb.y = -base.y;
    float bm = -dot64(base, m);  // dot(-base, m)
    float2 sub = mobius_pre(nb, m, b2, mm2, bm);
    float sub2 = dot64(sub, sub);
    float nrm = sqrtf(fmaxf(sub2, EPS));
    nrm = fminf(fmaxf(nrm, EPS), 1.f - 1e-5f);
    float sl = (2.f / lam) * atanhf(nrm) / nrm;

    float* orow = out_sum + (size_t)s * 64 + lane * 2;
    atomicAdd(orow, sub.x * sl);
    atomicAdd(orow + 1, sub.y * sl);
    if (lane == 0) atomicAdd(cnt1 + s, 1.f);
  }
}

__global__ void kg_norm_kernel(float* __restrict__ out,
                               const float* __restrict__ cnt1, size_t n64) {
  size_t i = (size_t)blockIdx.x * blockDim.x + threadIdx.x;
  size_t st = (size_t)gridDim.x * blockDim.x;
  for (; i < n64; i += st) out[i] /= fmaxf(cnt1[i >> 6], 1.f);
}

// Gate fusion via fp32 WMMA:
//   G = items @ W1^T + cf @ W2^T   (both accumulated into same C)
//   fusion = sigmoid(G)*items + (1-sigmoid(G))*cf
// Each wave owns ONE 16-col N-tile (tile = waveId & 3) so its weight (B)
// fragments for the whole K=64 are just 16 x v2f per matrix = 64 VGPRs,
// deliberately hoisted into registers across the strip loop. No LDS, no
// volatile, no scratch spills. 32 v_wmma per strip, fully unrolled.
__global__ void fusion_wmma_kernel(const float* __restrict__ ent,
                                   const float* __restrict__ cf,
                                   const float* __restrict__ W1,
                                   const float* __restrict__ W2,
                                   float* __restrict__ fusion, int n_items) {
  int lane = threadIdx.x & 31;
  int gwave = ((int)(blockIdx.x * blockDim.x + threadIdx.x)) >> 5;
  int nwaves = (int)((gridDim.x * blockDim.x) >> 5);
  int tile = gwave & 3;     // N-tile: cols [16*tile, 16*tile+16)
  int wstream = gwave >> 2; // strip stream id
  int nstreams = nwaves >> 2;
  int nstrips = n_items >> 4;

  int M = lane & 15;         // A row / B,C column within tile
  int kh = (lane >> 4) * 2;  // half-wave K offset (ISA 7.12.2, 32-bit 16x4)

  // Preload B fragments for this tile, all K, both matrices (register-resident)
  // B[k][n] = W[(16*tile+n)*64 + k]; lane -> n, VGPR pair -> K lsb, halfwave -> K+2
  const float* w1row = W1 + (size_t)(tile * 16 + M) * 64 + kh;
  const float* w2row = W2 + (size_t)(tile * 16 + M) * 64 + kh;
  v2f b1[16], b2[16];
#pragma unroll
  for (int k = 0; k < 16; k++) {
    b1[k][0] = w1row[k * 4];
    b1[k][1] = w1row[k * 4 + 1];
    b2[k][0] = w2row[k * 4];
    b2[k][1] = w2row[k * 4 + 1];
  }

  for (int strip = wstream; strip < nstrips; strip += nstreams) {
    int row0 = strip << 4;
    v8f acc = {};
    const float* ar = ent + (size_t)(row0 + M) * 64 + kh;
    const float* cr = cf + (size_t)(row0 + M) * 64 + kh;
#pragma unroll
    for (int k = 0; k < 16; k++) {
      v2f ai; ai[0] = ar[k * 4]; ai[1] = ar[k * 4 + 1];
      v2f ac; ac[0] = cr[k * 4]; ac[1] = cr[k * 4 + 1];
      acc = __builtin_amdgcn_wmma_f32_16x16x4_f32(
          false, ai, false, b1[k], (short)0, acc, false, false);
      acc = __builtin_amdgcn_wmma_f32_16x16x4_f32(
          false, ac, false, b2[k], (short)0, acc, false, false);
    }

    // Epilogue at C-layout positions: VGPR v -> row v + 8*(lane>=16),
    // lane&15 -> col within tile.
    int mofs = (lane >> 4) * 8;
    int col = tile * 16 + M;
#pragma unroll
    for (int v = 0; v < 8; v++) {
      size_t m = (size_t)(row0 + mofs + v);
      float it = ent[m * 64 + col];
      float cv = cf[m * 64 + col];
      float g = 1.f / (1.f + expf(-acc[v]));
      fusion[m * 64 + col] = g * it + (1.f - g) * cv;
    }
  }
}

// i2u aggregation: gather nodes[src] (item_fusion or user_embed), atomic
// segment-sum into d_out with the u/i_cf swap folded into the row mapping.
__global__ void i2u_kernel(const float* __restrict__ fusion,
                           const float* __restrict__ user_embed,
                           const int* __restrict__ src,
                           const int* __restrict__ dst,
                           float* __restrict__ out, float* __restrict__ cnt2,
                           int E, int n_items, int n_users, int N) {
  size_t i = (size_t)blockIdx.x * blockDim.x + threadIdx.x;
  size_t total = (size_t)E * 16;
  size_t st = (size_t)gridDim.x * blockDim.x;
  for (; i < total; i += st) {
    int e = (int)(i >> 4);
    int q = ((int)i & 15) * 4;
    int s = src[e];
    int d = dst[e];
    const float* srow = (s < n_items) ? (fusion + (size_t)s * 64)
                                      : (user_embed + (size_t)(s - n_items) * 64);
    float4 v = *(const float4*)(srow + q);
    // agg seg d -> d_out row: users first (u region), then items (i_cf region)
    size_t drow = (d < n_items) ? (size_t)(N + n_users + d)
                                : (size_t)(N + (d - n_items));
    float* o = out + drow * 64 + q;
    atomicAdd(o + 0, v.x);
    atomicAdd(o + 1, v.y);
    atomicAdd(o + 2, v.z);
    atomicAdd(o + 3, v.w);
    if (q == 0) atomicAdd(cnt2 + d, 1.f);
  }
}

__global__ void agg_norm_kernel(float* __restrict__ out,
                                const float* __restrict__ cnt2, int N,
                                int n_items, int n_users) {
  size_t total = (size_t)(n_items + n_users) * 64;
  size_t i = (size_t)blockIdx.x * blockDim.x + threadIdx.x;
  size_t st = (size_t)gridDim.x * blockDim.x;
  for (; i < total; i += st) {
    int r = (int)(i >> 6);  // 0..n_users-1 = u rows, then i_cf rows
    int s = (r < n_users) ? (n_items + r) : (r - n_users);
    out[(size_t)(N + r) * 64 + (i & 63)] /= fmaxf(cnt2[s], 1.f);
  }
}

extern "C" void kernel_launch(void* const* d_in, const int* in_sizes, int n_in,
                              void* d_out, int out_size, void* d_ws,
                              size_t ws_size, hipStream_t stream) {
  const float* entity_embed = (const float*)d_in[0];
  const float* user_embed   = (const float*)d_in[1];
  const float* relation_tab = (const float*)d_in[2];
  const float* item_cf      = (const float*)d_in[3];
  const float* gate1_W      = (const float*)d_in[4];
  const float* gate2_W      = (const float*)d_in[5];
  const int* edge_src  = (const int*)d_in[6];
  const int* edge_dst  = (const int*)d_in[7];
  const int* edge_type = (const int*)d_in[8];
  const int* i2u_src   = (const int*)d_in[9];
  const int* i2u_dst   = (const int*)d_in[10];

  const int N       = in_sizes[0] / 64;
  const int n_users = in_sizes[1] / 64;
  const int n_items = in_sizes[3] / 64;
  const int E_kg    = in_sizes[6];
  const int E_i2u   = in_sizes[9];

  float* out = (float*)d_out;

  // workspace layout (floats): [fusion n_items*64][cnt1 N][cnt2 n_items+n_users]
  float* ws_f   = (float*)d_ws;
  float* fusion = ws_f;
  float* cnt1   = fusion + (size_t)n_items * 64;
  float* cnt2   = cnt1 + (size_t)N;

  // 1) zero output + counters
  zero_kernel<<<4096, 256, 0, stream>>>(out, (size_t)out_size);
  zero_kernel<<<256, 256, 0, stream>>>(cnt1, (size_t)(N + n_items + n_users));

  // 2) KG edge pass (hyperbolic tangent-space messages + segment sums)
  kg_edge_kernel<<<8192, 256, 0, stream>>>(entity_embed, relation_tab, edge_src,
                                           edge_dst, edge_type, out, cnt1, E_kg);

  // 3) segment mean normalize for KG output
  kg_norm_kernel<<<8192, 256, 0, stream>>>(out, cnt1, (size_t)N * 64);

  // 4) gate fusion via fp32 WMMA: 4 waves (one per N-tile) x strip streams
  int nstrips = n_items >> 4;                 // 16-row strips
  int fwaves = nstrips * 4;                   // 4 tile-waves per strip stream
  int fblocks = (fwaves + 7) / 8;             // 8 waves per block
  fusion_wmma_kernel<<<fblocks, 256, 0, stream>>>(entity_embed, item_cf,
                                                  gate1_W, gate2_W, fusion,
                                                  n_items);

  // 5) i2u aggregation (atomic segment sums directly into d_out)
  i2u_kernel<<<8192, 256, 0, stream>>>(fusion, user_embed, i2u_src, i2u_dst,
                                       out, cnt2, E_i2u, n_items, n_users, N);

  // 6) segment mean normalize for u / i_cf regions
  agg_norm_kernel<<<4096, 256, 0, stream>>>(out, cnt2, N, n_items, n_users);
}